// ProtoHomophilyGate_824633721278
// MI455X (gfx1250) — compile-verified
//
#include <hip/hip_runtime.h>
#include <hip/hip_bf16.h>
#include <math.h>

typedef __attribute__((ext_vector_type(2))) float v2f;
typedef __attribute__((ext_vector_type(4))) float v4f;
typedef __attribute__((ext_vector_type(8))) float v8f;

#define D_FEAT 128
#define EPSV 1e-12f

// ---------------------------------------------------------------------------
// Kernel 1: one wave (32 lanes) per node.
//   rnorm[v] = 1 / max(||alpha[v]||, eps)
//   sim[v]   = self-loop cosine = ss * rnorm^2   (folds the self-loop edge)
//   cnt[v]   = 1.0                                (self-loop count)
// ---------------------------------------------------------------------------
__global__ void phg_norm_init(const float* __restrict__ alpha,
                              float* __restrict__ rnorm,
                              float* __restrict__ sim,
                              float* __restrict__ cnt,
                              int N) {
    int node = blockIdx.x * (blockDim.x >> 5) + (threadIdx.x >> 5);
    int lane = threadIdx.x & 31;
    if (node >= N) return;

    v4f v = *(const v4f*)(alpha + (size_t)node * D_FEAT + lane * 4);
    float ss = v.x * v.x + v.y * v.y + v.z * v.z + v.w * v.w;
    // wave32 butterfly reduction (all lanes end with the full sum)
    #pragma unroll
    for (int m = 16; m >= 1; m >>= 1) ss += __shfl_xor(ss, m, 32);

    if (lane == 0) {
        float n = sqrtf(ss);
        float d = fmaxf(n, EPSV);
        float r = 1.0f / d;
        rnorm[node] = r;
        sim[node]   = ss * r * r;  // dot(a_hat, a_hat) for the self loop
        cnt[node]   = 1.0f;
    }
}

// ---------------------------------------------------------------------------
// Kernel 2: one wave per group of 16 edges.
// Full-f32 WMMA (V_WMMA_F32_16X16X4_F32): A = 16 src-rows x 4K, B = 4K x 16
// tgt-cols, accumulate over K=128 in 32 chunks. Diagonal of the 16x16 result
// = the 16 edge dot-products. Diagonal e sits on lane e (e<8) / lane e+16
// (e>=8), vector component (lane & 7).
//
// A layout (16x4 f32): lane L<16 holds row M=L, {VGPR0,VGPR1} = K{0,1};
// lane L+16 holds K{2,3}. B (4x16) mirrors it with N in place of M.
// -> every lane loads a contiguous float2 at alpha[row]*128 + k0 + (lane>>4)*2.
// ---------------------------------------------------------------------------
__global__ void phg_edge_dot(const float* __restrict__ alpha,
                             const int* __restrict__ srcIdx,
                             const int* __restrict__ tgtIdx,
                             const float* __restrict__ rnorm,
                             float* __restrict__ sim,
                             float* __restrict__ cnt,
                             int E, int groups) {
    int g = blockIdx.x * (blockDim.x >> 5) + (threadIdx.x >> 5);
    int lane = threadIdx.x & 31;
    if (g >= groups) return;

    int e = g * 16 + (lane & 15);
    bool valid = (e < E);
    int s = valid ? srcIdx[e] : 0;
    int t = valid ? tgtIdx[e] : 0;

    int kk = (lane >> 4) * 2;  // K sub-offset within chunk: lanes 0-15 -> 0, 16-31 -> 2
    const float* arow = alpha + (size_t)s * D_FEAT + kk;
    const float* brow = alpha + (size_t)t * D_FEAT + kk;

    v8f c0 = {};
    v8f c1 = {};
    #pragma unroll
    for (int k0 = 0; k0 < D_FEAT; k0 += 8) {
        v2f a0 = *(const v2f*)(arow + k0);
        v2f b0 = *(const v2f*)(brow + k0);
        v2f a1 = *(const v2f*)(arow + k0 + 4);
        v2f b1 = *(const v2f*)(brow + k0 + 4);
        // D = A(16x4) * B(4x16) + C, full fp32 — matches reference precision
        c0 = __builtin_amdgcn_wmma_f32_16x16x4_f32(false, a0, false, b0,
                                                   (short)0, c0, false, false);
        c1 = __builtin_amdgcn_wmma_f32_16x16x4_f32(false, a1, false, b1,
                                                   (short)0, c1, false, false);
    }
    v8f c = c0 + c1;

    // Diagonal owners: lanes 0-7 (edges 0-7) and lanes 24-31 (edges 8-15)
    bool owner = (lane < 8) || (lane >= 24);
    float dotv = c[lane & 7];  // per-lane dynamic component select

    if (owner && valid) {
        float val = dotv * rnorm[s] * rnorm[t];  // cosine via factored norms
        atomicAdd(&sim[t], val);
        atomicAdd(&cnt[t], 1.0f);
    }
}

// ---------------------------------------------------------------------------
// Kernel 3: h[v] = sigmoid(T * sim[v] / max(cnt[v], 1))
// ---------------------------------------------------------------------------
__global__ void phg_finalize(const float* __restrict__ sim,
                             const float* __restrict__ cnt,
                             const float* __restrict__ temperature,
                             float* __restrict__ out,
                             int N) {
    int i = blockIdx.x * blockDim.x + threadIdx.x;
    if (i >= N) return;
    float mean = sim[i] / fmaxf(cnt[i], 1.0f);
    float x = temperature[0] * mean;
    out[i] = 1.0f / (1.0f + __expf(-x));
}

extern "C" void kernel_launch(void* const* d_in, const int* in_sizes, int n_in,
                              void* d_out, int out_size, void* d_ws, size_t ws_size,
                              hipStream_t stream) {
    const float* alpha       = (const float*)d_in[0];
    const float* temperature = (const float*)d_in[1];
    const int*   edge        = (const int*)d_in[2];  // harness: integer -> int32

    int N = in_sizes[0] / D_FEAT;
    int E = in_sizes[2] / 2;  // edge_index is [2, E]: row 0 = src, row 1 = tgt

    const int* srcIdx = edge;
    const int* tgtIdx = edge + E;

    float* ws    = (float*)d_ws;
    float* rnorm = ws;
    float* sim   = ws + N;
    float* cnt   = ws + 2 * N;
    float* out   = (float*)d_out;

    const int THREADS = 256;             // 8 waves per block (wave32)
    const int WPB     = THREADS / 32;

    int nBlocks = (N + WPB - 1) / WPB;
    phg_norm_init<<<nBlocks, THREADS, 0, stream>>>(alpha, rnorm, sim, cnt, N);

    int groups  = (E + 15) / 16;
    int eBlocks = (groups + WPB - 1) / WPB;
    phg_edge_dot<<<eBlocks, THREADS, 0, stream>>>(alpha, srcIdx, tgtIdx, rnorm,
                                                  sim, cnt, E, groups);

    int fBlocks = (N + THREADS - 1) / THREADS;
    phg_finalize<<<fBlocks, THREADS, 0, stream>>>(sim, cnt, temperature, out, N);
}